// TransformerBlock_970662608910
// MI455X (gfx1250) — compile-verified
//
#include <hip/hip_runtime.h>

// FAVOR+ causal linear attention (Performer), ReLU features, chunked parallel scan.
// B=2, L=4096, H=8, D=64, M=128. fp32 throughout, V_WMMA_F32_16X16X4_F32 tiles.
// Staging uses GLOBAL_LOAD_ASYNC_TO_LDS_B128 (ASYNCcnt) when the builtin exists.

#define Bn   2
#define Ln   4096
#define Hn   8
#define Dn   64
#define Mn   128
#define CHK  64
#define NC   (Ln / CHK)     // 64 chunks
#define BHn  (Bn * Hn)      // 16
#define DP   (Dn + 4)       // padded strides (64-bank LDS); 68 floats = 17*16B
#define MP   (Mn + 4)
#define CP   (CHK + 4)

#define RATIO 0.08838834764831845f  // 1/sqrt(128)
#define EPSF  1e-3f

// ---- LDS layout (floats), shared by pass A and pass C -----------------------
#define OFF_P   0
#define SZ_P    (Mn * DP)                 // 8704
#define OFF_Q   (OFF_P + SZ_P)            // q chunk; reused as masked A in pass C
#define SZ_CD   (CHK * DP)                // 4352
#define OFF_K   (OFF_Q + SZ_CD)
#define OFF_V   (OFF_K + SZ_CD)
#define OFF_QP  (OFF_V + SZ_CD)
#define SZ_CM   (CHK * MP)                // 8448
#define OFF_KP  (OFF_QP + SZ_CM)
#define OFF_Z   (OFF_KP + SZ_CM)
#define OFF_DEN (OFF_Z + Mn)
#define SMEM_FLOATS (OFF_DEN + CHK)       // 38848 floats
#define SMEM_BYTES  (SMEM_FLOATS * 4)     // 155392 B  (<160KB)

typedef __attribute__((ext_vector_type(2))) float v2f;
typedef __attribute__((ext_vector_type(8))) float v8f;
typedef __attribute__((ext_vector_type(4))) int   v4i;

#if defined(__has_builtin)
#if __has_builtin(__builtin_amdgcn_global_load_async_to_lds_b128)
#define ASYNC_LDS 1
#endif
#endif

// 16B global -> LDS copy; async (no VGPR round-trip, ASYNCcnt) when available.
__device__ __forceinline__ void cp16(const float* __restrict__ g, float* l) {
#ifdef ASYNC_LDS
  __builtin_amdgcn_global_load_async_to_lds_b128(
      (__attribute__((address_space(1))) v4i*)(v4i*)g,
      (__attribute__((address_space(3))) v4i*)(v4i*)l, 0, 0);
#else
  *(float4*)l = *(const float4*)g;
#endif
}

// Wait for this wave's async copies, then workgroup barrier.
__device__ __forceinline__ void async_join() {
#ifdef ASYNC_LDS
  asm volatile("s_wait_asynccnt 0" ::: "memory");
#endif
  __syncthreads();
}

// One 16x16 f32 tile, accumulating over K with V_WMMA_F32_16X16X4_F32.
// A element (r,k) at A[r*ars + k*acs]; B element (k,n) at Bm[k*brs + n*bcs].
__device__ __forceinline__ v8f wmma_acc_f32(const float* A, int ars, int acs,
                                            const float* Bm, int brs, int bcs,
                                            int K, v8f acc) {
  const int ln = threadIdx.x & 31;
  const int rc = ln & 15;           // A-row / B-col for this lane
  const int kh = (ln >> 4) << 1;    // K sub-offset: 0 or 2
  for (int k0 = 0; k0 < K; k0 += 4) {
    const int k = k0 + kh;
    v2f a, b;
    a.x = A[rc * ars + k * acs];
    a.y = A[rc * ars + (k + 1) * acs];
    b.x = Bm[k * brs + rc * bcs];
    b.y = Bm[(k + 1) * brs + rc * bcs];
    acc = __builtin_amdgcn_wmma_f32_16x16x4_f32(false, a, false, b,
                                                (short)0, acc, false, false);
  }
  return acc;
}

// ---- Pass A: per-chunk kp features + chunk states S_c = kp^T v, z_c --------
__global__ void favor_chunk_kv(const float* __restrict__ Kin,
                               const float* __restrict__ Vin,
                               const float* __restrict__ P,
                               float* __restrict__ Sc, float* __restrict__ Zc) {
  extern __shared__ float sm[];
  const int tid = threadIdx.x;
  const int c  = blockIdx.x % NC;
  const int bh = blockIdx.x / NC;
  const int b = bh / Hn, h = bh % Hn;
  const int l0 = c * CHK;

  float* Pb = sm + OFF_P;
  float* kb = sm + OFF_K;
  float* vb = sm + OFF_V;
  float* kp = sm + OFF_KP;

  // stage P (2048 x 16B granules) and k,v chunks (1024 granules each)
  for (int i = tid; i < (Mn * Dn) / 4; i += 256) {
    int m = i >> 4, d0 = (i & 15) << 2;
    cp16(P + m * Dn + d0, Pb + m * DP + d0);
  }
  for (int i = tid; i < (CHK * Dn) / 4; i += 256) {
    int t = i >> 4, d0 = (i & 15) << 2;
    size_t g = ((size_t)(b * Ln + l0 + t) * Hn + h) * Dn + d0;
    cp16(Kin + g, kb + t * DP + d0);
    cp16(Vin + g, vb + t * DP + d0);
  }
  async_join();

  const int w = tid >> 5;
  const int ln = tid & 31, col = ln & 15, rbase = (ln >> 4) << 3;

  // kp[t][m] = relu(RATIO * sum_d k[t][d] P[m][d]) + EPS   ([64 x 128])
  for (int ti = w; ti < (CHK / 16) * (Mn / 16); ti += 8) {
    int tr = ti / (Mn / 16), tc = ti % (Mn / 16);
    v8f acc = {};
    acc = wmma_acc_f32(kb + tr * 16 * DP, DP, 1,
                       Pb + tc * 16 * DP, 1, DP, Dn, acc);
    float* dst = kp + (tr * 16) * MP + tc * 16;
#pragma unroll
    for (int i = 0; i < 8; ++i)
      dst[(rbase + i) * MP + col] = fmaxf(acc[i] * RATIO, 0.0f) + EPSF;
  }
  __syncthreads();

  // S_c[m][d] = sum_t kp[t][m] v[t][d]   ([128 x 64])
  for (int ti = w; ti < (Mn / 16) * (Dn / 16); ti += 8) {
    int tr = ti / (Dn / 16), tc = ti % (Dn / 16);
    v8f acc = {};
    acc = wmma_acc_f32(kp + tr * 16, 1, MP,
                       vb + tc * 16, DP, 1, CHK, acc);
    size_t rowbase = (size_t)(bh * NC + c) * Mn + tr * 16;
#pragma unroll
    for (int i = 0; i < 8; ++i)
      Sc[(rowbase + rbase + i) * Dn + tc * 16 + col] = acc[i];
  }
  // z_c[m] = sum_t kp[t][m]
  if (tid < Mn) {
    float s = 0.0f;
    for (int t = 0; t < CHK; ++t) s += kp[t * MP + tid];
    Zc[(size_t)(bh * NC + c) * Mn + tid] = s;
  }
}

// ---- Pass B: exclusive prefix scan over chunks (element-parallel) ----------
__global__ void favor_scan_S(const float* __restrict__ Sc, float* __restrict__ Sp) {
  const int bh = blockIdx.x >> 5;                     // 32 blocks per head
  const int e = ((blockIdx.x & 31) << 8) + threadIdx.x;
  const size_t stride = (size_t)Mn * Dn;
  size_t base = (size_t)bh * NC * stride + e;
  float acc = 0.0f;
  for (int c = 0; c < NC; ++c) {
    Sp[base + (size_t)c * stride] = acc;
    acc += Sc[base + (size_t)c * stride];
  }
}

__global__ void favor_scan_Z(const float* __restrict__ Zc, float* __restrict__ Zp) {
  const int bh = blockIdx.x;
  const int m = threadIdx.x;                          // 128 threads
  size_t base = (size_t)bh * NC * Mn + m;
  float acc = 0.0f;
  for (int c = 0; c < NC; ++c) {
    Zp[base + (size_t)c * Mn] = acc;
    acc += Zc[base + (size_t)c * Mn];
  }
}

// ---- Pass C: qp/kp, intra-chunk causal A, out = A v + qp S_pre, /den -------
__global__ void favor_chunk_out(const float* __restrict__ Qin,
                                const float* __restrict__ Kin,
                                const float* __restrict__ Vin,
                                const float* __restrict__ P,
                                const float* __restrict__ Sp,
                                const float* __restrict__ Zp,
                                float* __restrict__ Out) {
  extern __shared__ float sm[];
  const int tid = threadIdx.x;
  const int c  = blockIdx.x % NC;
  const int bh = blockIdx.x / NC;
  const int b = bh / Hn, h = bh % Hn;
  const int l0 = c * CHK;

  float* Pb  = sm + OFF_P;    // P, later S_prefix tile
  float* qb  = sm + OFF_Q;    // q chunk, later masked A
  float* kb  = sm + OFF_K;
  float* vb  = sm + OFF_V;
  float* qp  = sm + OFF_QP;
  float* kp  = sm + OFF_KP;
  float* zb  = sm + OFF_Z;
  float* den = sm + OFF_DEN;

  for (int i = tid; i < (Mn * Dn) / 4; i += 256) {
    int m = i >> 4, d0 = (i & 15) << 2;
    cp16(P + m * Dn + d0, Pb + m * DP + d0);
  }
  for (int i = tid; i < (CHK * Dn) / 4; i += 256) {
    int t = i >> 4, d0 = (i & 15) << 2;
    size_t g = ((size_t)(b * Ln + l0 + t) * Hn + h) * Dn + d0;
    cp16(Qin + g, qb + t * DP + d0);
    cp16(Kin + g, kb + t * DP + d0);
    cp16(Vin + g, vb + t * DP + d0);
  }
  async_join();

  const int w = tid >> 5;
  const int ln = tid & 31, col = ln & 15, rbase = (ln >> 4) << 3;

  // qp and kp features: 64 tiles total over 8 waves
  for (int ti = w; ti < 2 * (CHK / 16) * (Mn / 16); ti += 8) {
    int which = ti >> 5;                 // 0 -> qp, 1 -> kp
    int t2 = ti & 31;
    int tr = t2 / (Mn / 16), tc = t2 % (Mn / 16);
    const float* src = which ? kb : qb;
    float* dst = (which ? kp : qp) + (tr * 16) * MP + tc * 16;
    v8f acc = {};
    acc = wmma_acc_f32(src + tr * 16 * DP, DP, 1,
                       Pb + tc * 16 * DP, 1, DP, Dn, acc);
#pragma unroll
    for (int i = 0; i < 8; ++i)
      dst[(rbase + i) * MP + col] = fmaxf(acc[i] * RATIO, 0.0f) + EPSF;
  }
  __syncthreads();

  // Kick off async reload of Pb with the S_prefix tile; it overlaps with the
  // entire A = qp kp^T GEMM below and is joined before the output GEMM.
  for (int i = tid; i < (Mn * Dn) / 4; i += 256) {
    int m = i >> 4, d0 = (i & 15) << 2;
    cp16(Sp + (size_t)(bh * NC + c) * Mn * Dn + m * Dn + d0, Pb + m * DP + d0);
  }
  if (tid < Mn) zb[tid] = Zp[(size_t)(bh * NC + c) * Mn + tid];

  // A = qp kp^T with inclusive causal mask, into qb region  ([64 x 64])
  float* Ab = qb;
  for (int ti = w; ti < (CHK / 16) * (CHK / 16); ti += 8) {
    int tr = ti / (CHK / 16), tc = ti % (CHK / 16);
    v8f acc = {};
    acc = wmma_acc_f32(qp + tr * 16 * MP, MP, 1,
                       kp + tc * 16, 1, MP, Mn, acc);
#pragma unroll
    for (int i = 0; i < 8; ++i) {
      int gr = tr * 16 + rbase + i;
      int gc = tc * 16 + col;
      Ab[gr * CP + gc] = (gc <= gr) ? acc[i] : 0.0f;
    }
  }
  async_join();   // A, S_prefix, z all resident

  // den[t] = rowsum(A_masked[t]) + qp[t] . z_prefix
  if (tid < CHK) {
    float s = 0.0f;
    for (int j = 0; j < CHK; ++j) s += Ab[tid * CP + j];
    for (int m = 0; m < Mn; ++m) s += qp[tid * MP + m] * zb[m];
    den[tid] = s;
  }
  __syncthreads();

  // out = A v + qp S_prefix, divided by den   ([64 x 64])
  for (int ti = w; ti < (CHK / 16) * (Dn / 16); ti += 8) {
    int tr = ti / (Dn / 16), tc = ti % (Dn / 16);
    v8f acc = {};
    acc = wmma_acc_f32(Ab + tr * 16 * CP, CP, 1,
                       vb + tc * 16, DP, 1, CHK, acc);
    acc = wmma_acc_f32(qp + tr * 16 * MP, MP, 1,
                       Pb + tc * 16, DP, 1, Mn, acc);
#pragma unroll
    for (int i = 0; i < 8; ++i) {
      int t = tr * 16 + rbase + i;
      float o = acc[i] / den[t];
      Out[((size_t)(b * Ln + l0 + t) * Hn + h) * Dn + tc * 16 + col] = o;
    }
  }
}

extern "C" void kernel_launch(void* const* d_in, const int* in_sizes, int n_in,
                              void* d_out, int out_size, void* d_ws, size_t ws_size,
                              hipStream_t stream) {
  const float* Q = (const float*)d_in[0];
  const float* K = (const float*)d_in[1];
  const float* V = (const float*)d_in[2];
  const float* P = (const float*)d_in[3];
  float* Out = (float*)d_out;

  // workspace: Sc | Sp | Zc | Zp
  const size_t szS = (size_t)BHn * NC * Mn * Dn * sizeof(float);  // 32 MB
  const size_t szZ = (size_t)BHn * NC * Mn * sizeof(float);       // 512 KB
  char* ws = (char*)d_ws;
  float* Sc = (float*)(ws);
  float* Sp = (float*)(ws + szS);
  float* Zc = (float*)(ws + 2 * szS);
  float* Zp = (float*)(ws + 2 * szS + szZ);

  (void)hipFuncSetAttribute((const void*)favor_chunk_kv,
                            hipFuncAttributeMaxDynamicSharedMemorySize, SMEM_BYTES);
  (void)hipFuncSetAttribute((const void*)favor_chunk_out,
                            hipFuncAttributeMaxDynamicSharedMemorySize, SMEM_BYTES);

  favor_chunk_kv<<<BHn * NC, 256, SMEM_BYTES, stream>>>(K, V, P, Sc, Zc);
  favor_scan_S<<<BHn * 32, 256, 0, stream>>>(Sc, Sp);
  favor_scan_Z<<<BHn, 128, 0, stream>>>(Zc, Zp);
  favor_chunk_out<<<BHn * NC, 256, SMEM_BYTES, stream>>>(Q, K, V, P, Sp, Zp, Out);
}